// DirectionalContrastiveLoss_88038239633768
// MI455X (gfx1250) — compile-verified
//
#include <hip/hip_runtime.h>
#include <math.h>

typedef __attribute__((ext_vector_type(2))) float v2f;
typedef __attribute__((ext_vector_type(8))) float v8f;

#define N_ROWS   16384
#define C_DIM    256
#define SEL      2000
#define M_COLS   4000          // 2*SEL == 125 * 32 exactly
#define HW       4096          // H*W = 64*64
#define INV_TEMP 10.0f
#define POS_TH   0.7f
#define EPS_F    1e-8f

#define WMMA_F32(a, b, c) \
  __builtin_amdgcn_wmma_f32_16x16x4_f32(false, (a), false, (b), (short)0, (c), false, false)

// ---- ordered-uint encoding of float so unsigned atomicMax == float max ----
__device__ __forceinline__ unsigned f2ord(float f) {
  unsigned u = __float_as_uint(f);
  return (u & 0x80000000u) ? ~u : (u | 0x80000000u);
}
__device__ __forceinline__ float ord2f(unsigned u) {
  return __uint_as_float((u & 0x80000000u) ? (u & 0x7fffffffu) : ~u);
}

// ---- 1) gather memory bank [M_COLS x C_DIM] from NCHW ul tensors ----------
__global__ void build_memory(const float* __restrict__ ul1,
                             const float* __restrict__ ul2,
                             const int* __restrict__ lab1,
                             const int* __restrict__ lab2,
                             const int* __restrict__ sel1,
                             const int* __restrict__ sel2,
                             float* __restrict__ mem,
                             int* __restrict__ memlab) {
  int m = blockIdx.x;          // 0..M_COLS-1
  int c = threadIdx.x;         // 0..C_DIM-1
  int idx; const float* src; const int* lab;
  if (m < SEL) { idx = sel1[m];       src = ul1; lab = lab1; }
  else         { idx = sel2[m - SEL]; src = ul2; lab = lab2; }
  int b  = idx >> 12;          // / HW
  int hw = idx & (HW - 1);
  // flat NCHW index: ((b*C + c)*H + h)*W + w == (b*C + c)*HW + hw
  mem[(size_t)m * C_DIM + c] = src[((size_t)(b * C_DIM + c)) * HW + hw];
  if (c == 0) memlab[m] = lab[idx];
}

// ---- 2) pos = dot(f1,f2)/T per row; init rowmax=pos, rowsum=0, scalars=0 --
__global__ void init_kernel(const float* __restrict__ f1,
                            const float* __restrict__ f2,
                            float* __restrict__ pos,
                            unsigned* __restrict__ rmax1,
                            unsigned* __restrict__ rmax2,
                            float* __restrict__ rsum1,
                            float* __restrict__ rsum2,
                            float* __restrict__ scal) {
  int row  = blockIdx.x * 8 + (threadIdx.x >> 5);   // wave per row
  int lane = threadIdx.x & 31;
  const float* a = f1 + (size_t)row * C_DIM;
  const float* b = f2 + (size_t)row * C_DIM;
  float acc = 0.f;
  #pragma unroll
  for (int c = lane; c < C_DIM; c += 32) acc += a[c] * b[c];
  #pragma unroll
  for (int off = 16; off >= 1; off >>= 1) acc += __shfl_xor(acc, off, 32);
  if (lane == 0) {
    float p = acc * INV_TEMP;
    pos[row] = p;
    unsigned e = f2ord(p);      // max starts at pos (pos column is unmasked)
    rmax1[row] = e;  rmax2[row] = e;
    rsum1[row] = 0.f; rsum2[row] = 0.f;
  }
  if (blockIdx.x == 0 && threadIdx.x < 4) scal[threadIdx.x] = 0.f;
}

// ---- epilogue helpers ------------------------------------------------------
// C/D layout: element r on lane (half,sub) = tile[M = r + 8*half][N = sub].
// ca/cb are the two 16-wide column tiles sharing the same 16 rows.
__device__ __forceinline__ void reduce_max_pair(v8f ca, v8f cb, int sub, int half,
                                                int rowbase,
                                                unsigned* __restrict__ rmax) {
  #pragma unroll
  for (int r = 0; r < 8; ++r) {
    float v = fmaxf(ca[r], cb[r]) * INV_TEMP;
    #pragma unroll
    for (int m = 8; m >= 1; m >>= 1) v = fmaxf(v, __shfl_xor(v, m, 32));
    if (sub == 0) atomicMax(rmax + rowbase + r + 8 * half, f2ord(v));
  }
}

__device__ __forceinline__ void reduce_sum_pair(v8f ca, v8f cb, int sub, int half,
                                                int rowbase, int mlab0, int mlab1,
                                                const int* __restrict__ lab,
                                                const unsigned* __restrict__ rmax,
                                                float* __restrict__ rsum) {
  #pragma unroll
  for (int r = 0; r < 8; ++r) {
    int row = rowbase + r + 8 * half;
    float mx = ord2f(rmax[row]);
    int   al = lab[row];
    float s = ((al != mlab0) ? __expf(ca[r] * INV_TEMP - mx) : 0.f)
            + ((al != mlab1) ? __expf(cb[r] * INV_TEMP - mx) : 0.f);
    #pragma unroll
    for (int m = 8; m >= 1; m >>= 1) s += __shfl_xor(s, m, 32);
    if (sub == 0) atomicAdd(rsum + row, s);
  }
}

// ---- 3) pass 1: 32x32 WMMA tile per wave (both branches), row max ---------
__global__ __launch_bounds__(32)
void gemm_max(const float* __restrict__ f1, const float* __restrict__ f2,
              const float* __restrict__ mem,
              unsigned* __restrict__ rmax1, unsigned* __restrict__ rmax2) {
  int n0   = blockIdx.x * 32;            // memory-column tile base
  int r0   = blockIdx.y * 32;            // anchor-row tile base
  int lane = threadIdx.x;                // wave32
  int half = lane >> 4;                  // selects K pair within 16x4 frag
  int sub  = lane & 15;                  // M (for A) / N (for B) index
  const float* A10 = f1  + (size_t)(r0 + sub) * C_DIM + 2 * half;
  const float* A11 = A10 + 16 * C_DIM;
  const float* A20 = f2  + (size_t)(r0 + sub) * C_DIM + 2 * half;
  const float* A21 = A20 + 16 * C_DIM;
  const float* B0  = mem + (size_t)(n0 + sub) * C_DIM + 2 * half;
  const float* B1  = B0  + 16 * C_DIM;

  v8f c1_00 = {}, c1_01 = {}, c1_10 = {}, c1_11 = {};  // branch1 [rt][ct]
  v8f c2_00 = {}, c2_01 = {}, c2_10 = {}, c2_11 = {};  // branch2 [rt][ct]
  #pragma unroll 4
  for (int k = 0; k < C_DIM; k += 4) {
    v2f a10 = *(const v2f*)(A10 + k);
    v2f a11 = *(const v2f*)(A11 + k);
    v2f a20 = *(const v2f*)(A20 + k);
    v2f a21 = *(const v2f*)(A21 + k);
    v2f b0  = *(const v2f*)(B0 + k);
    v2f b1  = *(const v2f*)(B1 + k);
    c1_00 = WMMA_F32(a10, b0, c1_00);
    c1_01 = WMMA_F32(a10, b1, c1_01);
    c1_10 = WMMA_F32(a11, b0, c1_10);
    c1_11 = WMMA_F32(a11, b1, c1_11);
    c2_00 = WMMA_F32(a20, b0, c2_00);
    c2_01 = WMMA_F32(a20, b1, c2_01);
    c2_10 = WMMA_F32(a21, b0, c2_10);
    c2_11 = WMMA_F32(a21, b1, c2_11);
  }
  reduce_max_pair(c1_00, c1_01, sub, half, r0,      rmax1);
  reduce_max_pair(c1_10, c1_11, sub, half, r0 + 16, rmax1);
  reduce_max_pair(c2_00, c2_01, sub, half, r0,      rmax2);
  reduce_max_pair(c2_10, c2_11, sub, half, r0 + 16, rmax2);
}

// ---- 4) pass 2: recompute tiles, masked exp-sum per row -------------------
__global__ __launch_bounds__(32)
void gemm_sum(const float* __restrict__ f1, const float* __restrict__ f2,
              const float* __restrict__ mem, const int* __restrict__ memlab,
              const int* __restrict__ lab1, const int* __restrict__ lab2,
              const unsigned* __restrict__ rmax1, const unsigned* __restrict__ rmax2,
              float* __restrict__ rsum1, float* __restrict__ rsum2) {
  int n0   = blockIdx.x * 32;
  int r0   = blockIdx.y * 32;
  int lane = threadIdx.x;
  int half = lane >> 4;
  int sub  = lane & 15;
  const float* A10 = f1  + (size_t)(r0 + sub) * C_DIM + 2 * half;
  const float* A11 = A10 + 16 * C_DIM;
  const float* A20 = f2  + (size_t)(r0 + sub) * C_DIM + 2 * half;
  const float* A21 = A20 + 16 * C_DIM;
  const float* B0  = mem + (size_t)(n0 + sub) * C_DIM + 2 * half;
  const float* B1  = B0  + 16 * C_DIM;

  v8f c1_00 = {}, c1_01 = {}, c1_10 = {}, c1_11 = {};
  v8f c2_00 = {}, c2_01 = {}, c2_10 = {}, c2_11 = {};
  #pragma unroll 4
  for (int k = 0; k < C_DIM; k += 4) {
    v2f a10 = *(const v2f*)(A10 + k);
    v2f a11 = *(const v2f*)(A11 + k);
    v2f a20 = *(const v2f*)(A20 + k);
    v2f a21 = *(const v2f*)(A21 + k);
    v2f b0  = *(const v2f*)(B0 + k);
    v2f b1  = *(const v2f*)(B1 + k);
    c1_00 = WMMA_F32(a10, b0, c1_00);
    c1_01 = WMMA_F32(a10, b1, c1_01);
    c1_10 = WMMA_F32(a11, b0, c1_10);
    c1_11 = WMMA_F32(a11, b1, c1_11);
    c2_00 = WMMA_F32(a20, b0, c2_00);
    c2_01 = WMMA_F32(a20, b1, c2_01);
    c2_10 = WMMA_F32(a21, b0, c2_10);
    c2_11 = WMMA_F32(a21, b1, c2_11);
  }
  int mlab0 = memlab[n0 + sub];        // label of this lane's column, ct=0
  int mlab1 = memlab[n0 + 16 + sub];   // label of this lane's column, ct=1
  reduce_sum_pair(c1_00, c1_01, sub, half, r0,      mlab0, mlab1, lab1, rmax1, rsum1);
  reduce_sum_pair(c1_10, c1_11, sub, half, r0 + 16, mlab0, mlab1, lab1, rmax1, rsum1);
  reduce_sum_pair(c2_00, c2_01, sub, half, r0,      mlab0, mlab1, lab2, rmax2, rsum2);
  reduce_sum_pair(c2_10, c2_11, sub, half, r0 + 16, mlab0, mlab1, lab2, rmax2, rsum2);
}

// ---- 5) per-row loss terms -> 4 scalars -----------------------------------
__global__ void finalize(const float* __restrict__ pos,
                         const unsigned* __restrict__ rmax1,
                         const unsigned* __restrict__ rmax2,
                         const float* __restrict__ rsum1,
                         const float* __restrict__ rsum2,
                         const float* __restrict__ plog1,
                         const float* __restrict__ plog2,
                         float* __restrict__ scal) {
  int i = blockIdx.x * blockDim.x + threadIdx.x;
  if (i >= N_ROWS) return;
  float p = pos[i];

  float m1 = ord2f(rmax1[i]);
  float e1 = __expf(p - m1);
  float lg1 = e1 / (e1 + rsum1[i] + EPS_F);
  float mk1 = (plog2[i] > POS_TH && plog1[i] < plog2[i]) ? 1.f : 0.f;

  float m2 = ord2f(rmax2[i]);
  float e2 = __expf(p - m2);
  float lg2 = e2 / (e2 + rsum2[i] + EPS_F);
  float mk2 = (plog1[i] > POS_TH && plog2[i] < plog1[i]) ? 1.f : 0.f;

  if (mk1 != 0.f) atomicAdd(&scal[0], -logf(lg1 + EPS_F));
  atomicAdd(&scal[1], mk1);
  if (mk2 != 0.f) atomicAdd(&scal[2], -logf(lg2 + EPS_F));
  atomicAdd(&scal[3], mk2);
}

__global__ void writeout(const float* __restrict__ scal, float* __restrict__ out) {
  out[0] = scal[0] / (scal[1] + 1e-12f) + scal[2] / (scal[3] + 1e-12f);
}

extern "C" void kernel_launch(void* const* d_in, const int* in_sizes, int n_in,
                              void* d_out, int out_size, void* d_ws, size_t ws_size,
                              hipStream_t stream) {
  const float* f1    = (const float*)d_in[0];
  const float* f2    = (const float*)d_in[1];
  const int*   lab1  = (const int*)  d_in[2];
  const int*   lab2  = (const int*)  d_in[3];
  const float* plog1 = (const float*)d_in[4];
  const float* plog2 = (const float*)d_in[5];
  const float* ul1   = (const float*)d_in[6];
  const float* ul2   = (const float*)d_in[7];
  const int*   sel1  = (const int*)  d_in[8];
  const int*   sel2  = (const int*)  d_in[9];

  // workspace layout (~4.5 MB)
  float*    mem    = (float*)d_ws;                        // M_COLS*C_DIM
  int*      memlab = (int*)(mem + (size_t)M_COLS * C_DIM);// M_COLS
  float*    pos    = (float*)(memlab + M_COLS);           // N_ROWS
  unsigned* rmax1  = (unsigned*)(pos + N_ROWS);           // N_ROWS
  unsigned* rmax2  = rmax1 + N_ROWS;                      // N_ROWS
  float*    rsum1  = (float*)(rmax2 + N_ROWS);            // N_ROWS
  float*    rsum2  = rsum1 + N_ROWS;                      // N_ROWS
  float*    scal   = rsum2 + N_ROWS;                      // 4

  build_memory<<<M_COLS, C_DIM, 0, stream>>>(ul1, ul2, lab1, lab2, sel1, sel2,
                                             mem, memlab);
  init_kernel<<<N_ROWS / 8, 256, 0, stream>>>(f1, f2, pos, rmax1, rmax2,
                                              rsum1, rsum2, scal);
  dim3 grid(M_COLS / 32, N_ROWS / 32);   // 125 x 512 waves per pass
  gemm_max<<<grid, 32, 0, stream>>>(f1, f2, mem, rmax1, rmax2);
  gemm_sum<<<grid, 32, 0, stream>>>(f1, f2, mem, memlab, lab1, lab2,
                                    rmax1, rmax2, rsum1, rsum2);
  finalize<<<N_ROWS / 256, 256, 0, stream>>>(pos, rmax1, rmax2, rsum1, rsum2,
                                             plog1, plog2, scal);
  writeout<<<1, 1, 0, stream>>>(scal, (float*)d_out);
}